// ProjectRenderScoreModel_85907935855038
// MI455X (gfx1250) — compile-verified
//
#include <hip/hip_runtime.h>

typedef float v2f __attribute__((ext_vector_type(2)));
typedef float v8f __attribute__((ext_vector_type(8)));

#define IMG_S   200
#define NPTS    128
#define NCAM    12          // BS*3
#define SSQ     40000       // 200*200
#define MASKS_N 480000      // 12*200*200
#define LOG2E   1.4426950408889634f
#define GSTEP   (2.0f / 199.0f)

__device__ __forceinline__ float flog2(float x) { return __builtin_amdgcn_logf(x); }
__device__ __forceinline__ float fexp2(float x) { return __builtin_amdgcn_exp2f(x); }
__device__ __forceinline__ float clip1(float x) { return fminf(fmaxf(x, -1.0f), 1.0f); }

// ---------------------------------------------------------------------------
// Kernel A: masks[b,cam,y,x] = min(max_n blob*inten, 1)
// Each wave handles 16 pixels; dst for a 16(point)x16(pixel) tile is computed
// with V_WMMA_F32_16X16X4_F32 using the exact rank-4 decomposition:
//   dst = (gx^2+gy^2)*1 + (-2gx)*px + (-2gy)*py + 1*(px^2+py^2)
// A-operand pairs are pre-packed in LDS per (K-half, point) so each tile needs
// only one uniform ds_load_b64 (no exec-divergent selection).
// ---------------------------------------------------------------------------
__global__ __launch_bounds__(128) void masks_wmma_kernel(
    const float* __restrict__ points2d,   // (12,128,2)
    const float* __restrict__ sigmas,     // (4,128)
    const float* __restrict__ exponents,  // (4,128)
    const float* __restrict__ intensities,// (4,128)
    const float* __restrict__ csig,       // (12)
    const float* __restrict__ cexp,       // (12)
    const float* __restrict__ cint,       // (12)
    float* __restrict__ masks_out)        // (12,200,200)
{
    __shared__ v2f   sA[2 * NPTS];        // [half*128+n] = A-operand pair
    __shared__ float se[NPTS], sc2[NPTS], sit[NPTS];

    const int c   = blockIdx.y;     // camera-flat index 0..11
    const int b   = c / 3;
    const int tid = threadIdx.x;    // 0..127

    // per-point parameter precompute (one point per thread)
    {
        const int n = tid;
        float px = points2d[(c * NPTS + n) * 2 + 0];
        float py = points2d[(c * NPTS + n) * 2 + 1];
        px = clip1((px - 100.0f) * 0.01f);
        py = clip1((py - 100.0f) * 0.01f);
        const float sg    = sigmas[b * NPTS + n] * csig[c];
        const float e     = exponents[b * NPTS + n] * cexp[c];
        const float it    = intensities[b * NPTS + n] * cint[c];
        const float inv2s = 0.5f / (sg * sg);       // 1/(2*sigma^2)
        v2f a0, a1;
        a0.x = 1.0f;  a0.y = px;                    // K0, K1 columns
        a1.x = py;    a1.y = px * px + py * py;     // K2, K3 columns
        sA[n]        = a0;
        sA[NPTS + n] = a1;
        se[n]  = e;
        sc2[n] = e * flog2(inv2s);                  // e*log2(1/(2s^2))
        sit[n] = it;
    }
    __syncthreads();

    const int lane  = tid & 31;
    const int wv    = tid >> 5;         // wave in block: 0..3
    const int half  = lane >> 4;        // K-half / M-half select
    const int l15   = lane & 15;
    const int pix   = blockIdx.x * 64 + wv * 16 + l15;   // pixel in this camera
    const int yi    = pix / IMG_S;
    const int xi    = pix - yi * IMG_S;
    const float gx  = fmaf((float)xi, GSTEP, -1.0f);
    const float gy  = fmaf((float)yi, GSTEP, -1.0f);

    // B operand (4x16, pixel-only terms) - invariant over n-tiles.
    // Layout: vB.x = B[half?2:0][l15], vB.y = B[half?3:1][l15]
    v2f bV;
    bV.x = half ? (-2.0f * gy) : fmaf(gx, gx, gy * gy);
    bV.y = half ? 1.0f         : (-2.0f * gx);

    const int aBase = half * NPTS + l15;    // LDS index base for A pairs

    float pm = 0.0f;
    #pragma unroll
    for (int t = 0; t < 8; ++t) {       // 8 tiles of 16 points
        const v2f aV = sA[aBase + t * 16];
        v8f cz = {};
        v8f d = __builtin_amdgcn_wmma_f32_16x16x4_f32(
            false, aV, false, bV, (short)0, cz, false, false);
        #pragma unroll
        for (int i = 0; i < 8; ++i) {
            const int r   = t * 16 + half * 8 + i;      // global point index
            const float dst  = fmaxf(d[i], 1e-20f);     // clamp fp cancellation
            const float L    = flog2(dst);
            const float u    = fexp2(fmaf(se[r], L, sc2[r]));   // t^e
            const float blob = fexp2(-LOG2E * u);               // exp(-t^e)
            pm = fmaxf(pm, blob * sit[r]);
        }
    }
    // lanes l and l+16 hold the same pixel (different point halves) -> merge
    pm = fmaxf(pm, __shfl_xor(pm, 16, 32));
    if (half == 0)
        masks_out[c * SSQ + pix] = fminf(pm, 1.0f);
}

// ---------------------------------------------------------------------------
// Kernel B: per (camera,n): peak = max_pix blob, wsum = sum_pix blob*target
// One block per (n, camera); deterministic LDS tree reduction, no atomics.
// ---------------------------------------------------------------------------
__global__ __launch_bounds__(256) void peak_wsum_kernel(
    const float* __restrict__ points2d,
    const float* __restrict__ sigmas,
    const float* __restrict__ exponents,
    const float* __restrict__ csig,
    const float* __restrict__ cexp,
    const float* __restrict__ target,   // (12,200,200)
    float* __restrict__ peakWS,         // (12,128)
    float* __restrict__ wsumWS)         // (12,128)
{
    const int n   = blockIdx.x;
    const int c   = blockIdx.y;
    const int b   = c / 3;
    const int tid = threadIdx.x;

    float px = points2d[(c * NPTS + n) * 2 + 0];
    float py = points2d[(c * NPTS + n) * 2 + 1];
    px = clip1((px - 100.0f) * 0.01f);
    py = clip1((py - 100.0f) * 0.01f);
    const float sg    = sigmas[b * NPTS + n] * csig[c];
    const float e     = exponents[b * NPTS + n] * cexp[c];
    const float inv2s = 0.5f / (sg * sg);
    const float c2    = e * flog2(inv2s);

    const float* tgt = target + c * SSQ;
    float pmax = 0.0f, ws = 0.0f;
    for (int p = tid; p < SSQ; p += 256) {
        const int   yi = p / IMG_S;
        const int   xi = p - yi * IMG_S;
        const float gx = fmaf((float)xi, GSTEP, -1.0f);
        const float gy = fmaf((float)yi, GSTEP, -1.0f);
        const float dx = gx - px, dy = gy - py;
        const float dst  = fmaxf(fmaf(dx, dx, dy * dy), 1e-20f);
        const float L    = flog2(dst);
        const float u    = fexp2(fmaf(e, L, c2));
        const float blob = fexp2(-LOG2E * u);
        pmax = fmaxf(pmax, blob);
        ws  += blob * tgt[p];
    }

    __shared__ float rm[256];
    __shared__ float rs[256];
    rm[tid] = pmax; rs[tid] = ws;
    __syncthreads();
    for (int s = 128; s > 0; s >>= 1) {
        if (tid < s) {
            rm[tid] = fmaxf(rm[tid], rm[tid + s]);
            rs[tid] += rs[tid + s];
        }
        __syncthreads();
    }
    if (tid == 0) {
        peakWS[c * NPTS + n] = fmaxf(rm[0], 1e-8f);
        wsumWS[c * NPTS + n] = rs[0];
    }
}

// ---------------------------------------------------------------------------
// Kernel C: scores[b,n] = mean_cam (wsum / peak)
// ---------------------------------------------------------------------------
__global__ void scores_kernel(const float* __restrict__ peakWS,
                              const float* __restrict__ wsumWS,
                              float* __restrict__ out)
{
    const int i = blockIdx.x * blockDim.x + threadIdx.x;
    if (i >= 4 * NPTS) return;
    const int b = i / NPTS;
    const int n = i - b * NPTS;
    float s = 0.0f;
    #pragma unroll
    for (int cam = 0; cam < 3; ++cam) {
        const int c = b * 3 + cam;
        s += wsumWS[c * NPTS + n] / peakWS[c * NPTS + n];
    }
    out[MASKS_N + i] = s * (1.0f / 3.0f);
}

// ---------------------------------------------------------------------------
extern "C" void kernel_launch(void* const* d_in, const int* in_sizes, int n_in,
                              void* d_out, int out_size, void* d_ws, size_t ws_size,
                              hipStream_t stream) {
    const float* points2d = (const float*)d_in[0];
    const float* sigmas   = (const float*)d_in[1];
    const float* expn     = (const float*)d_in[2];
    const float* inten    = (const float*)d_in[3];
    const float* csig     = (const float*)d_in[4];
    const float* cexp     = (const float*)d_in[5];
    const float* cint     = (const float*)d_in[6];
    const float* target   = (const float*)d_in[7];
    float* out = (float*)d_out;

    float* peakWS = (float*)d_ws;
    float* wsumWS = peakWS + NCAM * NPTS;

    dim3 gA(625, NCAM);   // 625*64 = 40000 pixels per camera
    masks_wmma_kernel<<<gA, 128, 0, stream>>>(points2d, sigmas, expn, inten,
                                              csig, cexp, cint, out);

    dim3 gB(NPTS, NCAM);
    peak_wsum_kernel<<<gB, 256, 0, stream>>>(points2d, sigmas, expn,
                                             csig, cexp, target,
                                             peakWS, wsumWS);

    scores_kernel<<<2, 256, 0, stream>>>(peakWS, wsumWS, out);
}